// HistogramMatchingLoss_6828998001492
// MI455X (gfx1250) — compile-verified
//
#include <hip/hip_runtime.h>
#include <hip/hip_bf16.h>

// MI455X / gfx1250, wave32.
// Pipeline: init ws -> minmax(pred) -> minmax(tgt) -> hist(pred) -> hist(tgt) -> mmd (WMMA f32 16x16x4)
// Roofline: two streaming passes over 160 MB => ~14 us at 23.3 TB/s; both arrays fit in 192 MB L2,
// so the histogram pass mostly hits L2. Final 64x64 MMD uses V_WMMA_F32_16X16X4_F32 (f32 precision).

typedef __attribute__((ext_vector_type(2))) float v2f;
typedef __attribute__((ext_vector_type(8))) float v8f;

#define NUM_BINS 64
#define WS_PMIN 0
#define WS_PMAX 1
#define WS_TMIN 2
#define WS_TMAX 3
#define WS_PHIST 4
#define WS_THIST (4 + NUM_BINS)
#define WS_TOTAL (4 + 2 * NUM_BINS)

// Monotone map: float -> u32 preserving order under unsigned compare.
__device__ __forceinline__ unsigned fkey(float f) {
    unsigned u = __float_as_uint(f);
    return (u & 0x80000000u) ? ~u : (u | 0x80000000u);
}
__device__ __forceinline__ float funkey(unsigned k) {
    unsigned u = (k & 0x80000000u) ? (k & 0x7FFFFFFFu) : ~k;
    return __uint_as_float(u);
}

__global__ void hm_init(unsigned* __restrict__ ws) {
    int i = threadIdx.x;
    if (i < WS_TOTAL) ws[i] = (i == WS_PMIN || i == WS_TMIN) ? 0xFFFFFFFFu : 0u;
}

__global__ __launch_bounds__(256) void hm_minmax(const float* __restrict__ x, int n,
                                                 unsigned* __restrict__ slots) {
    __shared__ unsigned smn[256], smx[256];
    unsigned kmn = 0xFFFFFFFFu, kmx = 0u;
    const int tid = threadIdx.x;
    const int gid = blockIdx.x * blockDim.x + tid;
    const int stride = gridDim.x * blockDim.x;
    const int n4 = n >> 2;
    const float4* __restrict__ x4 = (const float4*)x;
    for (int j = gid; j < n4; j += stride) {         // global_load_b128 stream
        float4 v = x4[j];
        unsigned a = fkey(v.x), b = fkey(v.y), c = fkey(v.z), d = fkey(v.w);
        kmn = min(kmn, min(min(a, b), min(c, d)));
        kmx = max(kmx, max(max(a, b), max(c, d)));
    }
    for (int j = (n4 << 2) + gid; j < n; j += stride) {
        unsigned a = fkey(x[j]);
        kmn = min(kmn, a);
        kmx = max(kmx, a);
    }
    smn[tid] = kmn; smx[tid] = kmx;
    __syncthreads();
    for (int off = 128; off > 0; off >>= 1) {
        if (tid < off) {
            smn[tid] = min(smn[tid], smn[tid + off]);
            smx[tid] = max(smx[tid], smx[tid + off]);
        }
        __syncthreads();
    }
    if (tid == 0) {
        atomicMin(&slots[0], smn[0]);
        atomicMax(&slots[1], smx[0]);
    }
}

__global__ __launch_bounds__(256) void hm_hist(const float* __restrict__ x, int n,
                                               const unsigned* __restrict__ slots,
                                               unsigned* __restrict__ gh) {
    // 8 wave32 per 256-thread block -> 8 privatized LDS histograms (2 KB of 320 KB).
    __shared__ unsigned h[8 * NUM_BINS];
    for (int k = threadIdx.x; k < 8 * NUM_BINS; k += blockDim.x) h[k] = 0;
    __syncthreads();
    const float mn = funkey(slots[0]);
    const float mx = funkey(slots[1]);
    float w = (mx - mn) * (1.0f / NUM_BINS);
    const float inv = (w > 0.f) ? 1.f / w : 0.f;
    unsigned* __restrict__ hw = &h[(threadIdx.x >> 5) * NUM_BINS];

    const int tid = threadIdx.x;
    const int gid = blockIdx.x * blockDim.x + tid;
    const int stride = gridDim.x * blockDim.x;
    const int n4 = n >> 2;
    const float4* __restrict__ x4 = (const float4*)x;
    for (int j = gid; j < n4; j += stride) {
        float4 v = x4[j];
        int b0 = (int)((v.x - mn) * inv); b0 = b0 < 0 ? 0 : (b0 > 63 ? 63 : b0);
        int b1 = (int)((v.y - mn) * inv); b1 = b1 < 0 ? 0 : (b1 > 63 ? 63 : b1);
        int b2 = (int)((v.z - mn) * inv); b2 = b2 < 0 ? 0 : (b2 > 63 ? 63 : b2);
        int b3 = (int)((v.w - mn) * inv); b3 = b3 < 0 ? 0 : (b3 > 63 ? 63 : b3);
        atomicAdd(&hw[b0], 1u);
        atomicAdd(&hw[b1], 1u);
        atomicAdd(&hw[b2], 1u);
        atomicAdd(&hw[b3], 1u);
    }
    for (int j = (n4 << 2) + gid; j < n; j += stride) {
        int b = (int)((x[j] - mn) * inv); b = b < 0 ? 0 : (b > 63 ? 63 : b);
        atomicAdd(&hw[b], 1u);
    }
    __syncthreads();
    for (int b = threadIdx.x; b < NUM_BINS; b += blockDim.x) {
        unsigned s = 0;
#pragma unroll
        for (int c = 0; c < 8; c++) s += h[c * NUM_BINS + b];
        if (s) atomicAdd(&gh[b], s);
    }
}

// Accumulate sum_{ij} f * W[i] * V[j] * exp(-(A[i]-B[j])^2/2) into acc via WMMA.
// A-operand layout (f32 16x4): lanes 0-15 hold K=0,1 in v0,v1; lanes 16-31 hold K=2,3.
// B operand is all-ones (layout-independent), so D[m][n] += sum_k A[m][k].
__device__ __forceinline__ v8f hm_accum_term(const float* A, const float* B,
                                             const float* W, const float* V,
                                             float f, int rlo, int khalf,
                                             v2f ones, v8f acc) {
    for (int I = 0; I < 4; I++) {
        const int row = I * 16 + rlo;
        const float av = A[row];
        const float wav = W[row] * f;
        for (int J = 0; J < 4; J++) {
            for (int c = 0; c < 4; c++) {
                v2f aop;
#pragma unroll
                for (int v = 0; v < 2; v++) {
                    const int col = J * 16 + c * 4 + khalf + v;
                    const float d = av - B[col];
                    aop[v] = wav * V[col] * expf(-d * d * 0.5f);  // SIGMA = 1
                }
                acc = __builtin_amdgcn_wmma_f32_16x16x4_f32(
                    false, aop, false, ones, (short)0, acc, false, false);
            }
        }
    }
    return acc;
}

// Final MMD over 64 bin centers via V_WMMA_F32_16X16X4_F32.
// One full wave32; EXEC all-1s around the WMMA ops (no divergence).
__global__ __launch_bounds__(32) void hm_mmd(const unsigned* __restrict__ ws, int np, int nt,
                                             float* __restrict__ out) {
    __shared__ float pc[NUM_BINS], tc[NUM_BINS], pw[NUM_BINS], tw[NUM_BINS];
    __shared__ float red[32];
    const int lane = threadIdx.x;

    const float pmin = funkey(ws[WS_PMIN]), pmax = funkey(ws[WS_PMAX]);
    const float tmin = funkey(ws[WS_TMIN]), tmax = funkey(ws[WS_TMAX]);
    float pwid = (pmax - pmin) * (1.0f / NUM_BINS); if (!(pwid > 0.f)) pwid = 1.f;
    float twid = (tmax - tmin) * (1.0f / NUM_BINS); if (!(twid > 0.f)) twid = 1.f;

    for (int i = lane; i < NUM_BINS; i += 32) {  // uniform trip count, no divergence
        float dens_p = (float)ws[WS_PHIST + i] / ((float)np * pwid);   // density=True
        pw[i] = fmaxf(1.f, floorf(dens_p * 100.f));
        pc[i] = pmin + pwid * ((float)i + 0.5f);
        float dens_t = (float)ws[WS_THIST + i] / ((float)nt * twid);
        tw[i] = fmaxf(1.f, floorf(dens_t * 100.f));
        tc[i] = tmin + twid * ((float)i + 0.5f);
    }
    __syncthreads();

    float Sp = 0.f, St = 0.f;
    for (int i = 0; i < NUM_BINS; i++) { Sp += pw[i]; St += tw[i]; }

    v8f acc = {};
    v2f ones; ones[0] = 1.0f; ones[1] = 1.0f;
    const int rlo = lane & 15;
    const int khalf = (lane >> 4) << 1;

    acc = hm_accum_term(pc, pc, pw, pw, 1.f / (Sp * Sp), rlo, khalf, ones, acc);
    acc = hm_accum_term(tc, tc, tw, tw, 1.f / (St * St), rlo, khalf, ones, acc);
    acc = hm_accum_term(pc, tc, pw, tw, -2.f / (Sp * St), rlo, khalf, ones, acc);

    // C/D layout: lanes 0-15 hold M=v (VGPR v), lanes 16-31 hold M=v+8; N = lane%16.
    // All columns of D are identical (B=ones) -> total = colsum(lane0) + colsum(lane16).
    float s = acc[0] + acc[1] + acc[2] + acc[3] + acc[4] + acc[5] + acc[6] + acc[7];
    red[lane] = s;
    __syncthreads();
    if (lane == 0) out[0] = red[0] + red[16];
}

extern "C" void kernel_launch(void* const* d_in, const int* in_sizes, int n_in,
                              void* d_out, int out_size, void* d_ws, size_t ws_size,
                              hipStream_t stream) {
    const float* p = (const float*)d_in[0];
    const float* t = (const float*)d_in[1];
    const int np = in_sizes[0];
    const int nt = in_sizes[1];
    unsigned* ws = (unsigned*)d_ws;
    float* out = (float*)d_out;

    hm_init<<<1, 256, 0, stream>>>(ws);
    const int blocks = 2048;  // grid-stride; enough waves to saturate 23.3 TB/s
    hm_minmax<<<blocks, 256, 0, stream>>>(p, np, ws + WS_PMIN);
    hm_minmax<<<blocks, 256, 0, stream>>>(t, nt, ws + WS_TMIN);
    hm_hist<<<blocks, 256, 0, stream>>>(p, np, ws + WS_PMIN, ws + WS_PHIST);
    hm_hist<<<blocks, 256, 0, stream>>>(t, nt, ws + WS_TMIN, ws + WS_THIST);
    hm_mmd<<<1, 32, 0, stream>>>(ws, np, nt, out);
}